// PDE2DReservoir_89661737271691
// MI455X (gfx1250) — compile-verified
//
#include <hip/hip_runtime.h>
#include <math.h>

#define NXg 4096
#define NYg 4096
#define STEPS 64
#define TILE 64
#define LSTRIDE 68   // LDS row stride in floats (64 interior + 2 halo + 2 pad, bank-friendly)

constexpr float DT_C   = 0.001f;
constexpr float DIFF_C = 0.01f;
constexpr float DX_C   = 0.02f;
constexpr float LAPC   = DT_C * DIFF_C / (DX_C * DX_C);   // 0.025

#if defined(__has_builtin)
#  if __has_builtin(__builtin_amdgcn_tensor_load_to_lds)
#    define HAS_TDM 1
#  endif
#  if __has_builtin(__builtin_amdgcn_global_load_async_to_lds_b32)
#    define HAS_ASYNC 1
#  endif
#endif
#ifndef HAS_TDM
#  define HAS_TDM 0
#endif
#ifndef HAS_ASYNC
#  define HAS_ASYNC 0
#endif

typedef unsigned int u32x4 __attribute__((ext_vector_type(4)));
typedef int          i32x8 __attribute__((ext_vector_type(8)));
typedef int          i32x4 __attribute__((ext_vector_type(4)));

typedef __attribute__((address_space(1))) int gint_t;   // global int
typedef __attribute__((address_space(3))) int lint_t;   // LDS int

// ---------------- zero U0 and forcing ----------------
__global__ void pde_zero2(float4* __restrict__ a, float4* __restrict__ b, int n4) {
    int i = blockIdx.x * blockDim.x + threadIdx.x;
    int stride = gridDim.x * blockDim.x;
    float4 z = make_float4(0.f, 0.f, 0.f, 0.f);
    for (; i < n4; i += stride) { a[i] = z; b[i] = z; }
}

// ---------------- scatter forcing field ----------------
__global__ void pde_scatter(const float* __restrict__ u_t,
                            const float* __restrict__ scales,
                            const int* __restrict__ dims,
                            const int* __restrict__ ix,
                            const int* __restrict__ iy,
                            float* __restrict__ F, int n) {
    int i = blockIdx.x * blockDim.x + threadIdx.x;
    if (i < n) {
        float v = scales[i] * u_t[dims[i]];
        atomicAdd(&F[(size_t)ix[i] * NYg + iy[i]], v);
    }
}

// ---------------- one diffusion step, TDM-staged tile in LDS ----------------
__global__ __launch_bounds__(256) void pde_step(const float* __restrict__ src,
                                                float* __restrict__ dst,
                                                const float* __restrict__ F,
                                                float coefF) {
    __shared__ float lds[(TILE + 2) * LSTRIDE];
    const int x0  = blockIdx.y * TILE;     // row-block (slow dim)
    const int y0  = blockIdx.x * TILE;     // col-block (contiguous dim)
    const int tid = threadIdx.x;

#if HAS_TDM
    // Wave 0 issues one Tensor-Data-Mover op: 64x64 f32 tile from global
    // into LDS interior, with pad (4 dwords after every 64) so the LDS
    // row stride lands on 68 floats.
    if (tid < 32) {
        unsigned long long ga =
            (unsigned long long)(const void*)(src + (size_t)x0 * NYg + y0);
        unsigned int lds_off = (unsigned int)(size_t)(&lds[1 * LSTRIDE + 1]);

        u32x4 g0;
        g0.x = 1u;                                            // count=1, load, no gather
        g0.y = lds_off;                                       // lds_addr (bytes)
        g0.z = (unsigned int)(ga & 0xFFFFFFFFull);            // global_addr[31:0]
        g0.w = (unsigned int)((ga >> 32) & 0x01FFFFFFull)     // global_addr[56:32]
             | (2u << 30);                                    // type = 2 ("image")

        i32x8 g1;
        g1[0] = (int)((2u << 16)      // data_size = 4B
                    | (1u << 20)      // pad_enable
                    | (5u << 22)      // pad_interval code 5 = 64 dwords
                    | (3u << 25));    // pad_amount  code 3 = 4 dwords
        g1[1] = (int)(((unsigned)NYg & 0xFFFFu) << 16);                    // tensor_dim0 lo16 @48
        g1[2] = (int)((((unsigned)NYg >> 16) & 0xFFFFu)
                    | (((unsigned)NXg & 0xFFFFu) << 16));                  // dim0 hi / dim1 lo
        g1[3] = (int)((((unsigned)NXg >> 16) & 0xFFFFu)
                    | ((unsigned)TILE << 16));                             // dim1 hi / tile_dim0
        g1[4] = (int)TILE;                                                 // tile_dim1, tile_dim2=0
        g1[5] = (int)NYg;                                                  // tensor_dim0_stride lo32
        g1[6] = 0;                                                         // stride0 hi / stride1 lo
        g1[7] = 0;
        i32x4 gz4 = {0, 0, 0, 0};
        i32x8 gz8 = {0, 0, 0, 0, 0, 0, 0, 0};
        // 6-arg form (clang-23 / therock-10.0 headers): extra int32x8 group, zero-filled.
        __builtin_amdgcn_tensor_load_to_lds(g0, g1, gz4, gz4, gz8, 0);
    }
#else
    // Fallback: cooperative interior load (16 coalesced elems/thread).
    for (int i = 0; i < 16; ++i) {
        int idx = i * 256 + tid;
        int x = idx >> 6, y = idx & 63;
        lds[(x + 1) * LSTRIDE + (y + 1)] = src[(size_t)(x0 + x) * NYg + (y0 + y)];
    }
#endif

    // Halo edges with periodic wrap (NX, NY are powers of two). 5-point
    // stencil needs no corners. One element per thread: compute the global
    // source element and the LDS destination element.
    {
        const int xm = (x0 - 1) & (NXg - 1);
        const int xp = (x0 + TILE) & (NXg - 1);
        const int ym = (y0 - 1) & (NYg - 1);
        const int yp = (y0 + TILE) & (NYg - 1);
        size_t gidx; int lidx;
        if (tid < 64) {                                   // top row
            gidx = (size_t)xm * NYg + (y0 + tid);
            lidx = 0 * LSTRIDE + 1 + tid;
        } else if (tid < 128) {                           // bottom row
            int j = tid - 64;
            gidx = (size_t)xp * NYg + (y0 + j);
            lidx = (TILE + 1) * LSTRIDE + 1 + j;
        } else if (tid < 192) {                           // left column
            int i = tid - 128;
            gidx = (size_t)(x0 + i) * NYg + ym;
            lidx = (i + 1) * LSTRIDE + 0;
        } else {                                          // right column
            int i = tid - 192;
            gidx = (size_t)(x0 + i) * NYg + yp;
            lidx = (i + 1) * LSTRIDE + (TILE + 1);
        }
#if HAS_ASYNC
        // Per-lane async global->LDS copy (ASYNCcnt path, no VGPR round-trip),
        // overlaps with the TDM interior DMA. Pointer casts go through
        // uintptr_t: AS1 = flat 64-bit VA, AS3 = low 32 bits (LDS offset).
        gint_t* gp = (gint_t*)(unsigned long long)(size_t)(src + gidx);
        lint_t* lp = (lint_t*)(unsigned int)(size_t)(&lds[lidx]);
        __builtin_amdgcn_global_load_async_to_lds_b32(gp, lp, 0, 0);
        asm volatile("s_wait_asynccnt 0" ::: "memory");
#else
        lds[lidx] = src[gidx];
#endif
    }

#if HAS_TDM
    if (tid < 32) __builtin_amdgcn_s_wait_tensorcnt(0);
#endif
    __syncthreads();

    // 256 threads x 16 iters cover the 64x64 tile; lanes walk contiguous y
    // (conflict-free LDS banks, coalesced stores).
    for (int i = 0; i < 16; ++i) {
        int x = i * 4 + (tid >> 6);
        int y = tid & 63;
        float c   = lds[(x + 1) * LSTRIDE + (y + 1)];
        float sum = lds[ x      * LSTRIDE + (y + 1)]
                  + lds[(x + 2) * LSTRIDE + (y + 1)]
                  + lds[(x + 1) * LSTRIDE +  y     ]
                  + lds[(x + 1) * LSTRIDE + (y + 2)]
                  - 4.0f * c;
        size_t g = (size_t)(x0 + x) * NYg + (y0 + y);
        dst[g] = c + LAPC * sum + coefF * F[g];
    }
}

// ---------------- gather measurements ----------------
__global__ void pde_gather(const float* __restrict__ U,
                           const int* __restrict__ ix,
                           const int* __restrict__ iy,
                           float* __restrict__ out, int n) {
    int i = blockIdx.x * blockDim.x + threadIdx.x;
    if (i < n) out[i] = U[(size_t)ix[i] * NYg + iy[i]];
}

extern "C" void kernel_launch(void* const* d_in, const int* in_sizes, int n_in,
                              void* d_out, int out_size, void* d_ws, size_t ws_size,
                              hipStream_t stream) {
    const float* u_t        = (const float*)d_in[0];
    const float* inj_scales = (const float*)d_in[1];
    const int*   inj_dims   = (const int*)d_in[2];
    const int*   inj_ix     = (const int*)d_in[3];
    const int*   inj_iy     = (const int*)d_in[4];
    const int*   meas_ix    = (const int*)d_in[5];
    const int*   meas_iy    = (const int*)d_in[6];
    float* out = (float*)d_out;

    const size_t NE = (size_t)NXg * NYg;
    float* Ua = (float*)d_ws;       // 64 MB
    float* Ub = Ua + NE;            // 64 MB
    float* F  = Ub + NE;            // 64 MB  (total 192 MB ~= L2 capacity)

    // U0 = 0, forcing = 0
    pde_zero2<<<4096, 256, 0, stream>>>((float4*)Ua, (float4*)F, (int)(NE / 4));

    // forcing[ix,iy] += scales * u_t[dims]
    const int nInj = in_sizes[1];
    pde_scatter<<<(nInj + 255) / 256, 256, 0, stream>>>(u_t, inj_scales, inj_dims,
                                                        inj_ix, inj_iy, F, nInj);

    // 64 ping-pong stencil steps; working set stays L2-resident
    dim3 grid(NYg / TILE, NXg / TILE);
    float* src = Ua;
    float* dst = Ub;
    for (int t = 0; t < STEPS; ++t) {
        float factor = sinf(3.14159265358979323846f * (float)t / 32.0f);
        pde_step<<<grid, 256, 0, stream>>>(src, dst, F, DT_C * factor);
        float* tmp = src; src = dst; dst = tmp;
    }

    // measured = U[meas_ix, meas_iy]  (after the final swap, result is in src)
    const int nMeas = out_size;
    pde_gather<<<(nMeas + 255) / 256, 256, 0, stream>>>(src, meas_ix, meas_iy, out, nMeas);
}